// Downsample_38671885533628
// MI455X (gfx1250) — compile-verified
//
#include <hip/hip_runtime.h>
#include <float.h>

typedef __attribute__((ext_vector_type(2))) float v2f;
typedef __attribute__((ext_vector_type(8))) float v8f;

// types for the gfx1250 async global->LDS builtin:
// expected param0: '__attribute__((__vector_size__(4*sizeof(int)))) int __device__ *'  (AS1, non-const)
typedef int v4i __attribute__((vector_size(16)));
typedef __attribute__((address_space(1))) v4i glob_v4i;
typedef __attribute__((address_space(3))) v4i lds_v4i;

#define S_LEN   500000
#define D_DIM   64
#define KSZ     5
#define NEW_S   249998        // (500000 - 5) / 2 + 1
#define TILE    128           // output rows per block
#define PAD     68            // LDS row stride in dwords (68 % 64 == 4 -> bank stepping, 16B-aligned rows)
#define NIN     (2*TILE + KSZ - 1)   // 260 input rows touched per block

__global__ __launch_bounds__(256) void downsample_wmma_kernel(
    const float* __restrict__ irreps,          // [S, 64]
    const float* __restrict__ coord,           // [S, 3]
    const unsigned char* __restrict__ mask_ir, // [S] bool
    const unsigned char* __restrict__ mask_co, // [S] bool
    const float* __restrict__ w_rel,           // [64]
    const float* __restrict__ w_out,           // [64, 64] row-major (d, e)
    float* __restrict__ out)                   // [N*64 | N*3 | N | N]
{
    __shared__ float s_in [NIN  * PAD];   // staged input tile (async copy target)
    __shared__ float s_avg[TILE * PAD];   // averaged window features (A matrix tile)
    __shared__ float s_wT [D_DIM * PAD];  // w_out transposed: s_wT[e*PAD + d] = w_out[d][e]
    __shared__ float s_s  [NIN + 1];      // per-input-row masked logit scalar
    __shared__ float s_mir[NIN + 1];
    __shared__ float s_mco[NIN + 1];

    const int tid = threadIdx.x;
    const long n0 = (long)blockIdx.x * TILE;

    // ---- Phase A: stage the 260x64 input tile into LDS (async DMA when available) ----
    for (int idx = tid; idx < NIN * 16; idx += 256) {
        const int row = idx >> 4;         // input row within tile
        const int c4  = idx & 15;         // float4 column
        const long gi = 2L * n0 + row;
        if (gi < S_LEN) {
            const float* gsrc = irreps + gi * D_DIM + 4 * c4;
            float*       ldst = s_in + row * PAD + 4 * c4;
#if __has_builtin(__builtin_amdgcn_global_load_async_to_lds_b128)
            glob_v4i* gp = (glob_v4i*)(unsigned long long)gsrc;  // generic/global share representation
            lds_v4i*  lp = (lds_v4i*)ldst;                        // addrspacecast -> LDS offset
            __builtin_amdgcn_global_load_async_to_lds_b128(gp, lp, 0, 0);
#else
            *(float4*)ldst = *(const float4*)gsrc;
#endif
        }
    }

    // ---- Phase 0: stage w_out^T into LDS (independent of the async tile) ----
    {
        const int e = tid & 63;
        for (int d = tid >> 6; d < D_DIM; d += 4)
            s_wT[e * PAD + d] = w_out[d * D_DIM + e];
    }

#if __has_builtin(__builtin_amdgcn_global_load_async_to_lds_b128)
#if __has_builtin(__builtin_amdgcn_s_wait_asynccnt)
    __builtin_amdgcn_s_wait_asynccnt(0);
#else
    asm volatile("s_wait_asynccnt 0" ::: "memory");
#endif
#endif
    __syncthreads();

    // ---- Phase 1: per-input-row scalar s[i] = m_ir * dot(irreps[i], w_rel), + masks ----
    for (int i = tid; i < NIN; i += 256) {
        const long gi = 2L * n0 + i;
        float mir = 0.f, mco = 0.f, sv = 0.f;
        if (gi < S_LEN) {
            mir = mask_ir[gi] ? 1.f : 0.f;
            mco = mask_co[gi] ? 1.f : 0.f;
            if (mir != 0.f) {
                const float4* row = (const float4*)(s_in + i * PAD);
                const float4* wr  = (const float4*)w_rel;
                float acc = 0.f;
#pragma unroll
                for (int j = 0; j < 16; ++j) {
                    const float4 v = row[j];
                    const float4 w = wr[j];
                    acc += v.x * w.x + v.y * w.y + v.z * w.z + v.w * w.w;
                }
                sv = acc;
            }
        }
        s_s[i] = sv; s_mir[i] = mir; s_mco[i] = mco;
    }
    __syncthreads();

    // ---- Phase 2: masked 5-tap mean (from LDS tile) -> avg tile in LDS (A matrix) ----
    {
        const int d4 = tid & 15;            // float4 column
        for (int r = tid >> 4; r < TILE; r += 16) {
            float4 acc = make_float4(0.f, 0.f, 0.f, 0.f);
            if (n0 + r < NEW_S) {
#pragma unroll
                for (int k = 0; k < KSZ; ++k) {
                    const int i = 2 * r + k;
                    if (s_mir[i] != 0.f) {
                        const float4 v = *(const float4*)(s_in + i * PAD + 4 * d4);
                        acc.x += v.x; acc.y += v.y; acc.z += v.z; acc.w += v.w;
                    }
                }
                acc.x *= 0.2f; acc.y *= 0.2f; acc.z *= 0.2f; acc.w *= 0.2f;
            }
            *(float4*)(s_avg + r * PAD + 4 * d4) = acc;
        }
    }

    // ---- Phase 3: softmax over K, coord output, masks (128 lanes, one row each) ----
    if (tid < TILE && (n0 + tid) < NEW_S) {
        const int r = tid;
        float l[KSZ], mc[KSZ];
        float mirsum = 0.f, mcosum = 0.f;
#pragma unroll
        for (int k = 0; k < KSZ; ++k) {
            const int i = 2 * r + k;
            mc[k]   = s_mco[i];
            mirsum += s_mir[i];
            mcosum += mc[k];
            l[k] = (mc[k] != 0.f) ? s_s[i] : -FLT_MAX;   // jnp.finfo.min == -FLT_MAX
        }
        float m = l[0];
#pragma unroll
        for (int k = 1; k < KSZ; ++k) m = fmaxf(m, l[k]);
        float e[KSZ], esum = 0.f;
#pragma unroll
        for (int k = 0; k < KSZ; ++k) { e[k] = __expf(l[k] - m); esum += e[k]; }
        const float inv = 1.f / esum;
        float cx = 0.f, cy = 0.f, cz = 0.f;
#pragma unroll
        for (int k = 0; k < KSZ; ++k) {
            const float w = e[k] * inv * mc[k];
            const float* cp = coord + (2L * (n0 + r) + k) * 3;
            cx += w * cp[0]; cy += w * cp[1]; cz += w * cp[2];
        }
        float* out_coord = out + (long)NEW_S * 64;
        out_coord[(n0 + r) * 3 + 0] = cx;
        out_coord[(n0 + r) * 3 + 1] = cy;
        out_coord[(n0 + r) * 3 + 2] = cz;
        out[(long)NEW_S * 67 + (n0 + r)] = (mirsum > 0.f) ? 1.f : 0.f;
        out[(long)NEW_S * 68 + (n0 + r)] = (mcosum > 0.f) ? 1.f : 0.f;
    }
    __syncthreads();

    // ---- Phase 4: per-wave 16x64 @ 64x64 GEMM via V_WMMA_F32_16X16X4_F32 ----
    // A frag (16x4 f32): lane 0-15 -> M, v0 = K(0|2), v1 = K(1|3) per lane half (ISA 7.12.2)
    // B frag (4x16 f32): lane 0-15 -> N, same K striping; sourced from w_out^T so pairs are contiguous
    {
        const int lane    = tid & 31;
        const int hi      = lane >> 4;        // lane half selects K+2 / M+8
        const int lrow    = lane & 15;
        const int rowBase = (tid >> 5) * 16;  // wave's 16 output rows within the tile

        v8f acc0 = {0.f,0.f,0.f,0.f,0.f,0.f,0.f,0.f};
        v8f acc1 = acc0, acc2 = acc0, acc3 = acc0;

#pragma unroll
        for (int kk = 0; kk < 16; ++kk) {          // 16 steps of K=4 over D=64
            const int kcol = 4 * kk + 2 * hi;       // even -> 8B-aligned LDS loads
            const v2f a  = *(const v2f*)(s_avg + (rowBase + lrow) * PAD + kcol);
            const v2f b0 = *(const v2f*)(s_wT  + ( 0 + lrow) * PAD + kcol);
            const v2f b1 = *(const v2f*)(s_wT  + (16 + lrow) * PAD + kcol);
            const v2f b2 = *(const v2f*)(s_wT  + (32 + lrow) * PAD + kcol);
            const v2f b3 = *(const v2f*)(s_wT  + (48 + lrow) * PAD + kcol);
            acc0 = __builtin_amdgcn_wmma_f32_16x16x4_f32(false, a, false, b0, (short)0, acc0, false, false);
            acc1 = __builtin_amdgcn_wmma_f32_16x16x4_f32(false, a, false, b1, (short)0, acc1, false, false);
            acc2 = __builtin_amdgcn_wmma_f32_16x16x4_f32(false, a, false, b2, (short)0, acc2, false, false);
            acc3 = __builtin_amdgcn_wmma_f32_16x16x4_f32(false, a, false, b3, (short)0, acc3, false, false);
        }

        // D layout: VGPR v holds row M = v + 8*hi, col N = lrow (+16 per tile)
#pragma unroll
        for (int v = 0; v < 8; ++v) {
            const long gr = n0 + rowBase + v + 8 * hi;
            if (gr < NEW_S) {
                float* o = out + gr * 64 + lrow;
                o[ 0] = acc0[v];
                o[16] = acc1[v];
                o[32] = acc2[v];
                o[48] = acc3[v];
            }
        }
    }
}

extern "C" void kernel_launch(void* const* d_in, const int* in_sizes, int n_in,
                              void* d_out, int out_size, void* d_ws, size_t ws_size,
                              hipStream_t stream) {
    const float*         irreps  = (const float*)d_in[0];
    const float*         coordp  = (const float*)d_in[1];
    const unsigned char* mask_ir = (const unsigned char*)d_in[2];
    const unsigned char* mask_co = (const unsigned char*)d_in[3];
    const float*         w_rel   = (const float*)d_in[4];
    const float*         w_out   = (const float*)d_in[5];
    float*               out     = (float*)d_out;

    const int blocks = (NEW_S + TILE - 1) / TILE;   // 1954
    downsample_wmma_kernel<<<blocks, 256, 0, stream>>>(
        irreps, coordp, mask_ir, mask_co, w_rel, w_out, out);
}